// MultiBoxLoss300_44160853738190
// MI455X (gfx1250) — compile-verified
//
#include <hip/hip_runtime.h>

#define NP    22536
#define NCLS  81
#define BATCH 32
#define NOBJ  16

typedef float v2f __attribute__((ext_vector_type(2)));
typedef float v8f __attribute__((ext_vector_type(8)));

__device__ __forceinline__ unsigned long long packIou(float iou, int p) {
    // iou >= 0 -> float bits monotonic as unsigned; ~p makes smaller prior index win ties
    return (((unsigned long long)__float_as_uint(iou)) << 32) | (unsigned int)(~(unsigned int)p);
}

__global__ void mb_init_kernel(unsigned long long* bestPacked) {
    int t = threadIdx.x;
    if (t < BATCH * NOBJ) bestPacked[t] = 0ULL;
}

// Per (b, prior): IoU against 16 boxes -> per-prior max/argmax; per (b,obj) argmax via packed u64 atomics.
__global__ void mb_match_kernel(const float4* __restrict__ priors,
                                const float4* __restrict__ boxes,
                                float* __restrict__ ov_prior,
                                int* __restrict__ obj_prior,
                                unsigned long long* __restrict__ bestPacked) {
    __shared__ float4 sbox[NOBJ];
    __shared__ float sarea[NOBJ];
    __shared__ unsigned long long smax[NOBJ];
    const int b = blockIdx.y;
    const int t = threadIdx.x;
    if (t < NOBJ) {
        float4 bb = boxes[b * NOBJ + t];
        sbox[t] = bb;
        sarea[t] = (bb.z - bb.x) * (bb.w - bb.y);
        smax[t] = 0ULL;
    }
    __syncthreads();
    const int p = blockIdx.x * blockDim.x + t;
    if (p < NP) {
        float4 pr = priors[p];
        float plx = pr.x - pr.z * 0.5f;
        float ply = pr.y - pr.w * 0.5f;
        float phx = pr.x + pr.z * 0.5f;
        float phy = pr.y + pr.w * 0.5f;
        float parea = (phx - plx) * (phy - ply);
        float best = -1.0f; int bo = 0;
        #pragma unroll
        for (int o = 0; o < NOBJ; ++o) {
            float4 bb = sbox[o];
            float ix = fminf(phx, bb.z) - fmaxf(plx, bb.x);
            float iy = fminf(phy, bb.w) - fmaxf(ply, bb.y);
            ix = fmaxf(ix, 0.0f); iy = fmaxf(iy, 0.0f);
            float inter = ix * iy;
            float iou = inter / (sarea[o] + parea - inter);
            if (iou > best) { best = iou; bo = o; }  // strict > = first-max, matches argmax(axis=0)
            atomicMax(&smax[o], packIou(iou, p));
        }
        ov_prior[(long long)b * NP + p] = best;
        obj_prior[(long long)b * NP + p] = bo;
    }
    __syncthreads();
    if (t < NOBJ) atomicMax(&bestPacked[b * NOBJ + t], smax[t]);
}

// Force each object's best prior: obj_prior[best]=o, ov_prior[best]=1.0 (sequential per b -> deterministic).
__global__ void mb_assign_kernel(const unsigned long long* __restrict__ bestPacked,
                                 float* __restrict__ ov_prior,
                                 int* __restrict__ obj_prior) {
    int b = threadIdx.x;
    if (b < BATCH) {
        for (int o = 0; o < NOBJ; ++o) {
            unsigned long long pk = bestPacked[b * NOBJ + o];
            int p = (int)(~(unsigned int)(pk & 0xFFFFFFFFULL));
            ov_prior[(long long)b * NP + p] = 1.0f;
            obj_prior[(long long)b * NP + p] = o;
        }
    }
}

// Main streaming kernel: focal log-softmax (81 classes) + DIoU on positives, WMMA wave reduction.
__global__ void mb_loss_kernel(const float4* __restrict__ locs,
                               const float* __restrict__ scores,
                               const float4* __restrict__ boxes,
                               const int* __restrict__ labels,
                               const float4* __restrict__ priors,
                               const float* __restrict__ ov_prior,
                               const int* __restrict__ obj_prior,
                               float* __restrict__ partials) {
    const int b = blockIdx.y;
    const int t = threadIdx.x;
    const int p = blockIdx.x * blockDim.x + t;
    float loc_s = 0.f, conf_s = 0.f, npos = 0.f, ninc = 0.f;
    if (p < NP) {
        const long long idx = (long long)b * NP + p;
        const float* sc = scores + idx * NCLS;
        __builtin_prefetch(sc, 0, 0);               // global_prefetch_b8: start score slab early
        __builtin_prefetch(sc + 32, 0, 0);
        __builtin_prefetch(sc + 64, 0, 0);
        float ovp = ov_prior[idx];
        int op = obj_prior[idx];
        int lab = labels[b * NOBJ + op];
        if (ovp < 0.5f) lab = -1;
        if (ovp < 0.4f) lab = 0;
        if (lab >= 0) {
            ninc = 1.f;
            float m = -3.4e38f, xt = 0.f;
            float v[NCLS];
            #pragma unroll
            for (int c = 0; c < NCLS; ++c) {
                float x = sc[c];
                v[c] = x;
                m = fmaxf(m, x);
                if (c == lab) xt = x;               // select, avoids dynamic register index
            }
            float s = 0.f;
            #pragma unroll
            for (int c = 0; c < NCLS; ++c) s += expf(v[c] - m);
            float logpt = xt - m - logf(s);
            float pt = expf(logpt);
            float om = 1.f - pt;
            conf_s = -(om * om) * logpt;            // focal, GAMMA=2
            if (lab > 0) {
                npos = 1.f;
                float4 g = locs[idx];
                float4 pr = priors[p];
                float cx = g.x * pr.z / 10.f + pr.x;
                float cy = g.y * pr.w / 10.f + pr.y;
                float w  = expf(g.z / 5.f) * pr.z;
                float h  = expf(g.w / 5.f) * pr.w;
                float px0 = cx - w * 0.5f, py0 = cy - h * 0.5f;
                float px1 = cx + w * 0.5f, py1 = cy + h * 0.5f;
                float4 tb = boxes[b * NOBJ + op];
                float ix = fminf(px1, tb.z) - fmaxf(px0, tb.x);
                float iy = fminf(py1, tb.w) - fmaxf(py0, tb.y);
                ix = fmaxf(ix, 0.f); iy = fmaxf(iy, 0.f);
                float inter = ix * iy;
                float ap = (px1 - px0) * (py1 - py0);
                float at = (tb.z - tb.x) * (tb.w - tb.y);
                float iou = inter / (ap + at - inter);
                float cpx = (px0 + px1) * 0.5f, cpy = (py0 + py1) * 0.5f;
                float ctx = (tb.x + tb.z) * 0.5f, cty = (tb.y + tb.w) * 0.5f;
                float dd = (cpx - ctx) * (cpx - ctx) + (cpy - cty) * (cpy - cty);
                float ox = fmaxf(px1, tb.z) - fminf(px0, tb.x);
                float oy = fmaxf(py1, tb.w) - fminf(py0, tb.y);
                ox = fmaxf(ox, 0.f); oy = fmaxf(oy, 0.f);
                float od = ox * ox + oy * oy;
                float d = iou - dd / od;
                d = fminf(fmaxf(d, -1.f), 1.f);
                loc_s = 1.f - d;
            }
        }
    }

    // ---- wave32 reduction via v_wmma_f32_16x16x4_f32 ----
    // A (selector): rows 0-7 = [1,0,1,0], rows 8-15 = [0,1,0,1]  (lane&8 picks which accumulator)
    // B (data): b.x/b.y per lane -> D row0 = x_n + x_{n+16}, row8 = y_n + y_{n+16}
    // D VGPR0: lanes 0-15 hold row0 (x pair-sums), lanes 16-31 hold row8 (y pair-sums)
    const int lane = t & 31;
    const int wave = t >> 5;
    const bool hi = (lane & 8) != 0;
    v2f asel = { hi ? 0.f : 1.f, hi ? 1.f : 0.f };
    v8f cz = {0.f, 0.f, 0.f, 0.f, 0.f, 0.f, 0.f, 0.f};

    v2f b1 = { loc_s, conf_s };
    v8f d1 = __builtin_amdgcn_wmma_f32_16x16x4_f32(false, asel, false, b1, (short)0, cz, false, false);
    float w1 = d1[0];
    w1 += __shfl_xor(w1, 1, 32);
    w1 += __shfl_xor(w1, 2, 32);
    w1 += __shfl_xor(w1, 4, 32);
    w1 += __shfl_xor(w1, 8, 32);   // lanes 0-15: sum(loc_s) over wave; lanes 16-31: sum(conf_s)

    v2f b2 = { npos, ninc };
    v8f d2 = __builtin_amdgcn_wmma_f32_16x16x4_f32(false, asel, false, b2, (short)0, cz, false, false);
    float w2 = d2[0];
    w2 += __shfl_xor(w2, 1, 32);
    w2 += __shfl_xor(w2, 2, 32);
    w2 += __shfl_xor(w2, 4, 32);
    w2 += __shfl_xor(w2, 8, 32);   // lanes 0-15: sum(npos); lanes 16-31: sum(ninc)

    __shared__ float part[8][4];
    if (lane == 0)  { part[wave][0] = w1; part[wave][2] = w2; }
    if (lane == 16) { part[wave][1] = w1; part[wave][3] = w2; }
    __syncthreads();
    if (t < 4) {
        float s = 0.f;
        #pragma unroll
        for (int w = 0; w < 8; ++w) s += part[w][t];
        partials[((long long)blockIdx.y * gridDim.x + blockIdx.x) * 4 + t] = s;
    }
}

__global__ void mb_final_kernel(const float* __restrict__ partials, int nblocks,
                                float* __restrict__ out) {
    __shared__ float r0[256], r1[256], r2[256], r3[256];
    const int t = threadIdx.x;
    float a0 = 0.f, a1 = 0.f, a2 = 0.f, a3 = 0.f;
    for (int i = t; i < nblocks; i += 256) {
        a0 += partials[(long long)i * 4 + 0];
        a1 += partials[(long long)i * 4 + 1];
        a2 += partials[(long long)i * 4 + 2];
        a3 += partials[(long long)i * 4 + 3];
    }
    r0[t] = a0; r1[t] = a1; r2[t] = a2; r3[t] = a3;
    for (int s = 128; s > 0; s >>= 1) {
        __syncthreads();
        if (t < s) { r0[t] += r0[t + s]; r1[t] += r1[t + s]; r2[t] += r2[t + s]; r3[t] += r3[t + s]; }
    }
    __syncthreads();
    if (t == 0) {
        float loc_sum = r0[0], conf_sum = r1[0], n_pos = r2[0], n_inc = r3[0];
        float conf_loss = (conf_sum / n_inc) / n_pos;
        float loc_loss  = loc_sum / n_pos;
        out[0] = conf_loss + 25.0f * loc_loss;   // ALPHA = 25
    }
}

extern "C" void kernel_launch(void* const* d_in, const int* in_sizes, int n_in,
                              void* d_out, int out_size, void* d_ws, size_t ws_size,
                              hipStream_t stream) {
    const float4* locs   = (const float4*)d_in[0];   // [32, 22536, 4] f32
    const float*  scores = (const float*)d_in[1];    // [32, 22536, 81] f32
    const float4* boxes  = (const float4*)d_in[2];   // [32, 16, 4] f32
    const int*    labels = (const int*)d_in[3];      // [32, 16] i32
    const float4* priors = (const float4*)d_in[4];   // [22536, 4] f32
    float* out = (float*)d_out;

    char* ws = (char*)d_ws;
    unsigned long long* bestPacked = (unsigned long long*)ws;              // 512 * 8B
    size_t off = 4096;
    float* ov_prior = (float*)(ws + off); off += (size_t)BATCH * NP * 4;   // 2.88 MB
    int*   obj_prior = (int*)(ws + off);  off += (size_t)BATCH * NP * 4;   // 2.88 MB
    float* partials = (float*)(ws + off);                                  // 2848 * 16B

    const int gx = (NP + 255) / 256;   // 89

    mb_init_kernel<<<1, 512, 0, stream>>>(bestPacked);
    mb_match_kernel<<<dim3(gx, BATCH), 256, 0, stream>>>(priors, boxes, ov_prior, obj_prior, bestPacked);
    mb_assign_kernel<<<1, 32, 0, stream>>>(bestPacked, ov_prior, obj_prior);
    mb_loss_kernel<<<dim3(gx, BATCH), 256, 0, stream>>>(locs, scores, boxes, labels, priors,
                                                        ov_prior, obj_prior, partials);
    mb_final_kernel<<<1, 256, 0, stream>>>(partials, gx * BATCH, out);
}